// ts_corr_47596827574813
// MI455X (gfx1250) — compile-verified
//
#include <hip/hip_runtime.h>
#include <stdint.h>

typedef __attribute__((ext_vector_type(2))) float v2f;
typedef __attribute__((ext_vector_type(8))) float v8f;

#define NCH   32
#define TLEN  240
#define DWIN  10
#define NW    24
#define NPAIR 496   // 32*31/2

// linear index of pair (i,j), i<j, in row-major upper-triangle (k=1) order
__device__ __forceinline__ int triu_index(int i, int j) {
    return i * 31 - (i * (i - 1)) / 2 + (j - i - 1);
}

__global__ __launch_bounds__(256) void ts_corr_wmma_kernel(
        const float* __restrict__ X, float* __restrict__ out) {
    // 32 x 240 f32 slab for this batch; normalized in place (windows tile T exactly)
    __shared__ float lds[NCH * TLEN];  // 30720 B of the 320 KB WGP LDS

    const int b   = blockIdx.x;
    const int tid = threadIdx.x;

    // ---- Phase 1: async global -> LDS copy (GLOBAL_LOAD_ASYNC_TO_LDS_B64) ----
    // 30720 B = 15 iterations * 256 lanes * 8 B. ASYNCcnt-tracked; no VGPR bounce.
    {
        const uint32_t lds_base = (uint32_t)(uintptr_t)(&lds[0]);  // low 32 bits = LDS offset
        const uint64_t gbase    = (uint64_t)(uintptr_t)(X + (size_t)b * (NCH * TLEN));
        #pragma unroll
        for (int k = 0; k < (NCH * TLEN) / (256 * 2); ++k) {
            const uint32_t byte_off = (uint32_t)(k * 256 + tid) * 8u;
            const uint32_t dst      = lds_base + byte_off;
            const uint64_t src      = gbase + byte_off;
            asm volatile("global_load_async_to_lds_b64 %0, %1, off"
                         :: "v"(dst), "v"(src) : "memory");
        }
        asm volatile("s_wait_asynccnt 0x0" ::: "memory");
    }
    __syncthreads();

    // ---- Phase 2: normalize each (n,w) window in place. 768 tasks, 3/thread ----
    #pragma unroll
    for (int k = 0; k < 3; ++k) {
        const int task = k * 256 + tid;
        const int n = task / NW;
        const int w = task % NW;
        float* p = &lds[n * TLEN + w * DWIN];
        float v[DWIN];
        float s = 0.f;
        #pragma unroll
        for (int d = 0; d < DWIN; ++d) { v[d] = p[d]; s += v[d]; }
        const float mean = s * (1.0f / DWIN);
        float ss = 0.f;
        #pragma unroll
        for (int d = 0; d < DWIN; ++d) { v[d] -= mean; ss += v[d] * v[d]; }
        const float inv = rsqrtf(ss);
        #pragma unroll
        for (int d = 0; d < DWIN; ++d) p[d] = v[d] * inv;
    }
    __syncthreads();

    // ---- Phase 3: per-wave Gram via V_WMMA_F32_16X16X4_F32 (3 windows/wave) ----
    const int wave = tid >> 5;
    const int lane = tid & 31;
    const int r0   = lane & 15;        // A row / B col, tile 0
    const int r1   = r0 + 16;          // A row / B col, tile 1
    const int kh   = (lane >> 4) << 1; // K sub-offset: 0 (lanes 0-15) or 2 (lanes 16-31)
    const bool hi  = (lane >= 16);

    float* __restrict__ outb = out + (size_t)b * (NPAIR * NW);

    for (int wi = 0; wi < 3; ++wi) {
        const int w = wave * 3 + wi;
        const float* __restrict__ x0 = &lds[r0 * TLEN + w * DWIN];
        const float* __restrict__ x1 = &lds[r1 * TLEN + w * DWIN];

        v8f c00 = {};  // rows 0-15  x rows 0-15
        v8f c01 = {};  // rows 0-15  x rows 16-31
        v8f c11 = {};  // rows 16-31 x rows 16-31

        // K chunks 0,1: all lanes in range (k <= 7 < 10) -> unconditional LDS loads
        #pragma unroll
        for (int c = 0; c < 2; ++c) {
            const int k0 = c * 4 + kh;
            v2f a0 = { x0[k0], x0[k0 + 1] };
            v2f a1 = { x1[k0], x1[k0 + 1] };
            c00 = __builtin_amdgcn_wmma_f32_16x16x4_f32(false, a0, false, a0,
                                                        (short)0, c00, false, false);
            c01 = __builtin_amdgcn_wmma_f32_16x16x4_f32(false, a0, false, a1,
                                                        (short)0, c01, false, false);
            c11 = __builtin_amdgcn_wmma_f32_16x16x4_f32(false, a1, false, a1,
                                                        (short)0, c11, false, false);
        }

        // K chunk 2: lanes 0-15 supply k={8,9}; lanes 16-31 are zero pad (k=10,11).
        // Clamp index so the load is unconditional, then cndmask the value.
        {
            const int kc = hi ? 0 : 8;                 // safe in-bounds index
            const float t00 = x0[kc], t01 = x0[kc + 1];
            const float t10 = x1[kc], t11 = x1[kc + 1];
            v2f a0 = { hi ? 0.f : t00, hi ? 0.f : t01 };
            v2f a1 = { hi ? 0.f : t10, hi ? 0.f : t11 };
            c00 = __builtin_amdgcn_wmma_f32_16x16x4_f32(false, a0, false, a0,
                                                        (short)0, c00, false, false);
            c01 = __builtin_amdgcn_wmma_f32_16x16x4_f32(false, a0, false, a1,
                                                        (short)0, c01, false, false);
            c11 = __builtin_amdgcn_wmma_f32_16x16x4_f32(false, a1, false, a1,
                                                        (short)0, c11, false, false);
        }

        // C/D f32 layout: lane L, VGPR g -> M = g + 8*(L>=16), N = L&15
        const int Mbase = (lane >> 4) << 3;
        const int N     = lane & 15;
        #pragma unroll
        for (int g = 0; g < 8; ++g) {
            const int M = Mbase + g;
            if (M < N) outb[triu_index(M, N) * NW + w] = c00[g];           // T00 upper
            outb[triu_index(M, N + 16) * NW + w] = c01[g];                 // T01 full
            if (M < N) outb[triu_index(M + 16, N + 16) * NW + w] = c11[g]; // T11 upper
        }
    }
}

extern "C" void kernel_launch(void* const* d_in, const int* in_sizes, int n_in,
                              void* d_out, int out_size, void* d_ws, size_t ws_size,
                              hipStream_t stream) {
    (void)n_in; (void)out_size; (void)d_ws; (void)ws_size;
    const float* X = (const float*)d_in[0];
    float* out     = (float*)d_out;
    const int B    = in_sizes[0] / (NCH * TLEN);   // 2048
    ts_corr_wmma_kernel<<<B, 256, 0, stream>>>(X, out);
}